// LinearGatedProto_8083128451615
// MI455X (gfx1250) — compile-verified
//
#include <hip/hip_runtime.h>

typedef float v2f  __attribute__((ext_vector_type(2)));
typedef float v8f  __attribute__((ext_vector_type(8)));
typedef unsigned int u32x4 __attribute__((ext_vector_type(4)));
typedef int   i32x4 __attribute__((ext_vector_type(4)));
typedef int   i32x8 __attribute__((ext_vector_type(8)));
typedef __attribute__((address_space(3))) float lds_f;

#define N_ 8192
#define I_ 1024
#define O_ 1024
#define P_ 8
#define C_ 64
#define M_ (O_ * P_)   // 8192

#define BN 128          // n rows per workgroup
#define BM 128          // m cols per workgroup (16 o-columns)
#define KB 16           // K chunk (floats)
#define LSTR (KB + 4)   // LDS row stride in floats == TDM pad (16 dw + 4 dw pad)
#define TILE_F (BN * LSTR)

__global__ __launch_bounds__(256)
void fused_proto_gemm(const float* __restrict__ X,
                      const float* __restrict__ ctx,
                      const float* __restrict__ W,
                      const float* __restrict__ bias,
                      const float* __restrict__ proto,
                      float* __restrict__ out)
{
    // 4 tile buffers (A0,B0,A1,B1) + proto_sq[BM]
    __shared__ float lds_mem[4 * TILE_F + BM];
    float* psq = lds_mem + 4 * TILE_F;
    const unsigned lds_base = (unsigned)(unsigned long long)(lds_f*)lds_mem;

    const int tid  = threadIdx.x;
    const int lane = tid & 31;
    const int wid  = tid >> 5;
    const int hf   = lane >> 4;          // half-wave: K pair select / M+8 row
    const int r16  = lane & 15;
    const int m_wave = (wid & 3) * 32;   // 4 waves along m, 32 m each
    const int n_wave = (wid >> 2) * 64;  // 2 waves along n, 64 n each
    const int m0 = blockIdx.x * BM;
    const int n0 = blockIdx.y * BN;

    // ---- proto_sq[m] = sum_c proto[m,c]^2 into LDS ----
    {
        const int r = tid >> 1, h = tid & 1;
        const float* pr = proto + (size_t)(m0 + r) * C_ + h * 32;
        float s = 0.f;
        #pragma unroll
        for (int i = 0; i < 32; i += 4) {
            float4 v = *(const float4*)(pr + i);
            s += v.x * v.x + v.y * v.y + v.z * v.z + v.w * v.w;
        }
        lds_mem[tid] = s;
        __syncthreads();
        if (tid < BM) psq[tid] = lds_mem[2 * tid] + lds_mem[2 * tid + 1];
        __syncthreads();
    }

    // ---- TDM: DMA one 128 x KB f32 tile (row stride ld) into LDS with
    //      pad_interval=16 DWORDs, pad_amount=4 DWORDs  ->  LDS stride 20 ----
    auto tdm_issue = [&](const float* g, unsigned ld, unsigned ldsOffBytes) {
        unsigned long long ga = (unsigned long long)(uintptr_t)g;
        u32x4 g0;
        g0[0] = 1u;                                   // count=1 (valid), user mode
        g0[1] = lds_base + ldsOffBytes;               // lds_addr (bytes)
        g0[2] = (unsigned)ga;                         // global_addr[31:0]
        g0[3] = (unsigned)((ga >> 32) & 0x01FFFFFFull) | 0x80000000u; // [56:32] | type=2
        i32x8 g1;
        g1[0] = (2 << 16)    // data_size = 4B
              | (1 << 20)    // pad_enable
              | (3 << 22)    // pad_interval: 16 DWORDs
              | (3 << 25);   // pad_amount:   4 DWORDs
        g1[1] = (int)((ld & 0xFFFFu) << 16);              // tensor_dim0 lo16
        g1[2] = (int)((ld >> 16) | ((unsigned)BN << 16)); // dim0 hi | tensor_dim1=BN
        g1[3] = (int)(KB << 16);                          // dim1 hi=0 | tile_dim0=KB
        g1[4] = BN;                                       // tile_dim1=BN, tile_dim2=0
        g1[5] = (int)ld;                                  // tensor_dim0_stride lo32
        g1[6] = 0;                                        // stride hi | dim1_stride lo
        g1[7] = 0;
        i32x4 gz4 = (i32x4)0;
        i32x8 gz8 = (i32x8)0;
        __builtin_amdgcn_tensor_load_to_lds(g0, g1, gz4, gz4, gz8, 0);
    };

    // ---- TDM-fed, double-buffered WMMA GEMM phase ----
    auto phase = [&](const float* gA, int ldA, const float* gB, int ldB,
                     int kTot, v8f (&acc)[4][2]) {
        const int nCh = kTot / KB;
        if (wid == 0) {
            tdm_issue(gA, ldA, 0 * TILE_F * 4);
            tdm_issue(gB, ldB, 1 * TILE_F * 4);
        }
        for (int kc = 0; kc < nCh; ++kc) {
            const int buf = kc & 1;
            if (wid == 0) {
                if (kc + 1 < nCh) {
                    tdm_issue(gA + (kc + 1) * KB, ldA, ((buf ^ 1) * 2 + 0) * TILE_F * 4);
                    tdm_issue(gB + (kc + 1) * KB, ldB, ((buf ^ 1) * 2 + 1) * TILE_F * 4);
                    __builtin_amdgcn_s_wait_tensorcnt(2);  // chunk kc complete
                } else {
                    __builtin_amdgcn_s_wait_tensorcnt(0);
                }
            }
            __syncthreads();
            const float* A = lds_mem + (buf * 2 + 0) * TILE_F;
            const float* B = lds_mem + (buf * 2 + 1) * TILE_F;
            #pragma unroll
            for (int ks = 0; ks < KB; ks += 4) {
                v2f af[4], bf[2];
                #pragma unroll
                for (int t = 0; t < 4; ++t)
                    af[t] = *(const v2f*)(A + (n_wave + t * 16 + r16) * LSTR + ks + 2 * hf);
                #pragma unroll
                for (int u = 0; u < 2; ++u)
                    bf[u] = *(const v2f*)(B + (m_wave + u * 16 + r16) * LSTR + ks + 2 * hf);
                #pragma unroll
                for (int t = 0; t < 4; ++t)
                    #pragma unroll
                    for (int u = 0; u < 2; ++u)
                        acc[t][u] = __builtin_amdgcn_wmma_f32_16x16x4_f32(
                            false, af[t], false, bf[u], (short)0, acc[t][u],
                            false, false);
            }
            __syncthreads();
        }
    };

    v8f acc[4][2];
    #pragma unroll
    for (int t = 0; t < 4; ++t)
        #pragma unroll
        for (int u = 0; u < 2; ++u) acc[t][u] = (v8f)0.f;

    // ---- Phase 1: cross = ctx * proto^T (K = C), reduce to packed argmin p ----
    phase(ctx + (size_t)n0 * C_, C_, proto + (size_t)m0 * C_, C_, C_, acc);

    unsigned pse[8];   // pse[t*2+u]: 8 x 4-bit winning p (one nibble per r)
    #pragma unroll
    for (int t = 0; t < 4; ++t) {
        #pragma unroll
        for (int u = 0; u < 2; ++u) {
            const float ps = psq[m_wave + u * 16 + r16];
            unsigned pack = 0;
            #pragma unroll
            for (int r = 0; r < 8; ++r) {
                // d2 = proto_sq - 2*cross (+ ctx_sq[n], const per n -> dropped)
                float s = ps - 2.0f * acc[t][u][r];
                int bp = lane & 7;
                #pragma unroll
                for (int off = 1; off < 8; off <<= 1) {
                    float os = __shfl_xor(s, off, 8);
                    int   op = __shfl_xor(bp, off, 8);
                    if (os < s || (os == s && op < bp)) { s = os; bp = op; }
                }
                pack |= (unsigned)bp << (r * 4);
            }
            pse[t * 2 + u] = pack;
        }
    }

    // ---- Phase 2: all_out = X * W^T (K = I), reusing the same accumulators ----
    #pragma unroll
    for (int t = 0; t < 4; ++t)
        #pragma unroll
        for (int u = 0; u < 2; ++u) acc[t][u] = (v8f)0.f;

    phase(X + (size_t)n0 * I_, I_, W + (size_t)m0 * I_, I_, I_, acc);

    // ---- Finalize: bias add, gather winner across the 8-lane p-group, store ----
    #pragma unroll
    for (int t = 0; t < 4; ++t) {
        #pragma unroll
        for (int u = 0; u < 2; ++u) {
            const float bv = bias[m0 + m_wave + u * 16 + r16];
            const unsigned pack = pse[t * 2 + u];
            #pragma unroll
            for (int r = 0; r < 8; ++r) {
                float val = acc[t][u][r] + bv;
                int bp = (int)((pack >> (r * 4)) & 7u);
                float sel = __shfl(val, bp, 8);   // winner's (biased) output
                if ((lane & 7) == 0) {
                    int n = n0 + n_wave + t * 16 + r + 8 * hf;
                    int o = (m0 + m_wave + u * 16 + r16) >> 3;
                    out[(size_t)n * O_ + o] = sel;
                }
            }
        }
    }
}

extern "C" void kernel_launch(void* const* d_in, const int* in_sizes, int n_in,
                              void* d_out, int out_size, void* d_ws, size_t ws_size,
                              hipStream_t stream) {
    const float* X     = (const float*)d_in[0];
    const float* ctx   = (const float*)d_in[1];
    const float* W     = (const float*)d_in[2];
    const float* bias  = (const float*)d_in[3];
    const float* proto = (const float*)d_in[4];
    float* out = (float*)d_out;

    dim3 grid(M_ / BM, N_ / BN);   // 64 x 64 workgroups
    fused_proto_gemm<<<grid, dim3(256), 0, stream>>>(X, ctx, W, bias, proto, out);
}